// SheafWindowLayer_36507222016689
// MI455X (gfx1250) — compile-verified
//
#include <hip/hip_runtime.h>
#include <hip/hip_bf16.h>
#include <math.h>

typedef __attribute__((ext_vector_type(2))) float v2f;
typedef __attribute__((ext_vector_type(8))) float v8f;

#define B_DIM 2
#define L_DIM 1024
#define D_DIM 768
#define K_WIN 64
#define H_DIM 128
#define W_DIM (L_DIM - K_WIN + 1)   /* 961  */
#define NROWS (B_DIM * W_DIM)       /* 1922 */
#define M2PAD 1952                  /* 61 * 32 */
#define EPS_F 1e-8f

// ---------------------------------------------------------------------------
// fp32 WMMA GEMM, 2x2 tiles (32x32 output) per wave32:
//   C[M x N] = A[M x Kd] * B[Kd x N]
// V_WMMA_F32_16X16X4_F32 along K. A-row pointers for rows >= Mvalid are
// clamped (safe reads); since D row m depends only on A row m, the garbage
// rows are simply not stored (r < Mvalid guard). No masking in the hot loop.
// ---------------------------------------------------------------------------
__global__ __launch_bounds__(32)
void wmma_gemm_f32_kernel(const float* __restrict__ A,
                          const float* __restrict__ B,
                          float* __restrict__ C,
                          int N, int Kd, int Mvalid)
{
    const int m0   = blockIdx.x << 5;    // 32-row macro tile
    const int n0   = blockIdx.y << 5;    // 32-col macro tile
    const int lane = threadIdx.x & 31;
    const int lo   = lane & 15;
    const int hi   = lane >> 4;          // 0: K 0,1 | 1: K 2,3 (A and B)
    const int koff = hi << 1;

    // A-matrix 16x4 f32 layout (ISA 7.12.2): lanes 0-15 = rows M, VGPR0/1 = K pair.
    const int rowA0 = m0 + lo;
    const int rowA1 = m0 + 16 + lo;
    const int rA0c  = (rowA0 < Mvalid) ? rowA0 : (Mvalid - 1);
    const int rA1c  = (rowA1 < Mvalid) ? rowA1 : (Mvalid - 1);
    const float* Arow0 = A + (size_t)rA0c * (size_t)Kd + koff;
    const float* Arow1 = A + (size_t)rA1c * (size_t)Kd + koff;

    // B-matrix 4x16 f32 layout: lanes = N columns, VGPR v holds K = koff + v.
    const float* Bp = B + (size_t)koff * (size_t)N + n0 + lo;

    v8f acc00 = {0.f,0.f,0.f,0.f,0.f,0.f,0.f,0.f};
    v8f acc01 = acc00, acc10 = acc00, acc11 = acc00;

    for (int k0 = 0; k0 < Kd; k0 += 4) {
        v2f a0 = *(const v2f*)(Arow0 + k0);   // 8B-aligned (k0+koff even)
        v2f a1 = *(const v2f*)(Arow1 + k0);
        v2f b0, b1;
        const float* brow0 = Bp + (size_t)k0 * N;
        const float* brow1 = Bp + (size_t)(k0 + 1) * N;
        b0.x = brow0[0];   b0.y = brow1[0];
        b1.x = brow0[16];  b1.y = brow1[16];
        __builtin_prefetch(Arow0 + k0 + 32, 0, 1);   // global_prefetch_b8
        __builtin_prefetch(Arow1 + k0 + 32, 0, 1);
        acc00 = __builtin_amdgcn_wmma_f32_16x16x4_f32(false, a0, false, b0, (short)0, acc00, false, false);
        acc01 = __builtin_amdgcn_wmma_f32_16x16x4_f32(false, a0, false, b1, (short)0, acc01, false, false);
        acc10 = __builtin_amdgcn_wmma_f32_16x16x4_f32(false, a1, false, b0, (short)0, acc10, false, false);
        acc11 = __builtin_amdgcn_wmma_f32_16x16x4_f32(false, a1, false, b1, (short)0, acc11, false, false);
    }

    // C/D 16x16 f32 layout: VGPR v -> row v (lanes 0-15) / row v+8 (lanes 16-31)
#pragma unroll
    for (int v = 0; v < 8; ++v) {
        const int r0 = m0 + v + (hi << 3);
        const int r1 = r0 + 16;
        if (r0 < Mvalid) {
            C[(size_t)r0 * (size_t)N + n0 + lo]      = acc00[v];
            C[(size_t)r0 * (size_t)N + n0 + 16 + lo] = acc01[v];
        }
        if (r1 < Mvalid) {
            C[(size_t)r1 * (size_t)N + n0 + lo]      = acc10[v];
            C[(size_t)r1 * (size_t)N + n0 + 16 + lo] = acc11[v];
        }
    }
}

// ---------------------------------------------------------------------------
// Hbar[row, h] = (1/63) * sum_{k=0..62} relu(Y[b, w+k, h] + CC[k,h] + b1[h])
// row = b*W + w.  Rows >= NROWS (padding) are written as zero.
// ---------------------------------------------------------------------------
__global__ __launch_bounds__(H_DIM)
void window_relu_mean_kernel(const float* __restrict__ Y,    // (B*L, H)
                             const float* __restrict__ CC,   // (64, H), rows 0..62 valid
                             const float* __restrict__ b1,   // (H)
                             float* __restrict__ Hbar)       // (M2PAD, H)
{
    const int h   = threadIdx.x;
    const int row = blockIdx.x;
    if (row >= NROWS) { Hbar[(size_t)row * H_DIM + h] = 0.f; return; }
    const int b = row / W_DIM;
    const int w = row % W_DIM;
    const float* y = Y + ((size_t)(b * L_DIM + w)) * H_DIM + h;
    const float bias = b1[h];
    float s = 0.f;
#pragma unroll 7
    for (int k = 0; k < K_WIN - 1; ++k) {
        float v = y[(size_t)k * H_DIM] + CC[(size_t)k * H_DIM + h] + bias;
        s += (v > 0.f) ? v : 0.f;
    }
    Hbar[(size_t)row * H_DIM + h] = s * (1.0f / (float)(K_WIN - 1));
}

// ---------------------------------------------------------------------------
// Epilogue: per window row, form mean_delta / update, write x_out row,
// and emit per-row stats (deterministic; no atomics).
//   rs[row] = { sum(delta^2), alpha*sqrt(sum(delta^2)), cos_sim(row) }
// ---------------------------------------------------------------------------
__global__ __launch_bounds__(256)
void epilogue_kernel(const float* __restrict__ x,
                     const float* __restrict__ MD0,     // (M2PAD, D) = Hbar*W2
                     const float* __restrict__ b2,
                     const float* __restrict__ alpha_p,
                     float* __restrict__ x_out,
                     float* __restrict__ rs)            // (M2PAD, 4)
{
    const int row = blockIdx.x;                 // 0..NROWS-1
    const int b   = row / W_DIM;
    const int w   = row % W_DIM;
    const int t   = w + K_WIN - 1;
    const float alpha = *alpha_p;

    const float* xc = x     + ((size_t)(b * L_DIM + t)) * D_DIM;
    const float* md = MD0   + (size_t)row * D_DIM;
    float*       xo = x_out + ((size_t)(b * L_DIM + t)) * D_DIM;

    float s1 = 0.f, s2 = 0.f, s3 = 0.f, s4 = 0.f;
    for (int d = threadIdx.x; d < D_DIM; d += 256) {
        float cur   = xc[d];
        float delta = cur - (md[d] + b2[d]);    // mean_delta
        float nb    = cur - alpha * delta;      // cur + update
        xo[d] = nb;
        s1 += delta * delta;
        s2 += cur * nb;
        s3 += cur * cur;
        s4 += nb * nb;
    }

    // wave32 shuffle reduce, then cross-wave via LDS
#pragma unroll
    for (int off = 16; off > 0; off >>= 1) {
        s1 += __shfl_down(s1, off, 32);
        s2 += __shfl_down(s2, off, 32);
        s3 += __shfl_down(s3, off, 32);
        s4 += __shfl_down(s4, off, 32);
    }
    __shared__ float sm[8][4];
    const int wid = threadIdx.x >> 5;
    const int ln  = threadIdx.x & 31;
    if (ln == 0) { sm[wid][0] = s1; sm[wid][1] = s2; sm[wid][2] = s3; sm[wid][3] = s4; }
    __syncthreads();
    if (threadIdx.x == 0) {
        float t1 = 0.f, t2 = 0.f, t3 = 0.f, t4 = 0.f;
#pragma unroll
        for (int i = 0; i < 8; ++i) { t1 += sm[i][0]; t2 += sm[i][1]; t3 += sm[i][2]; t4 += sm[i][3]; }
        float na = fmaxf(sqrtf(t3), EPS_F);
        float nb = fmaxf(sqrtf(t4), EPS_F);
        rs[(size_t)row * 4 + 0] = t1;
        rs[(size_t)row * 4 + 1] = fabsf(alpha) * sqrtf(t1);
        rs[(size_t)row * 4 + 2] = t2 / (na * nb);
    }
}

// Copy the first K-1 rows per batch (untouched prefix of x_out).
__global__ __launch_bounds__(256)
void copy_head_kernel(const float* __restrict__ x, float* __restrict__ x_out)
{
    const int n   = B_DIM * (K_WIN - 1) * D_DIM;       // 96768
    const int idx = blockIdx.x * 256 + threadIdx.x;
    if (idx >= n) return;
    const int b = idx / ((K_WIN - 1) * D_DIM);
    const int r = idx % ((K_WIN - 1) * D_DIM);
    x_out[(size_t)b * L_DIM * D_DIM + r] = x[(size_t)b * L_DIM * D_DIM + r];
}

// Deterministic fixed-order reduction of per-row stats -> the 4 scalar outputs.
__global__ __launch_bounds__(256)
void finalize_kernel(const float* __restrict__ rs,
                     const float* __restrict__ alpha_p,
                     float* __restrict__ out_tail)      // 4 floats
{
    __shared__ float sm[256][3];
    float a0 = 0.f, a1 = 0.f, a2 = 0.f;
    for (int row = threadIdx.x; row < NROWS; row += 256) {
        a0 += rs[(size_t)row * 4 + 0];
        a1 += rs[(size_t)row * 4 + 1];
        a2 += rs[(size_t)row * 4 + 2];
    }
    sm[threadIdx.x][0] = a0; sm[threadIdx.x][1] = a1; sm[threadIdx.x][2] = a2;
    __syncthreads();
    for (int s = 128; s > 0; s >>= 1) {
        if (threadIdx.x < s) {
            sm[threadIdx.x][0] += sm[threadIdx.x + s][0];
            sm[threadIdx.x][1] += sm[threadIdx.x + s][1];
            sm[threadIdx.x][2] += sm[threadIdx.x + s][2];
        }
        __syncthreads();
    }
    if (threadIdx.x == 0) {
        const float alpha = *alpha_p;
        const float inv   = 1.0f / (float)NROWS;
        float pre = sm[0][0] * inv;
        out_tail[0] = pre;                                   // first_pre_energy
        out_tail[1] = (1.0f - alpha) * (1.0f - alpha) * pre; // final_post_energy
        out_tail[2] = sm[0][1] * inv;                        // total_update_norm / DIFF_STEPS
        out_tail[3] = sm[0][2] * inv;                        // total_cos_sim / DIFF_STEPS
    }
}

// ---------------------------------------------------------------------------
extern "C" void kernel_launch(void* const* d_in, const int* in_sizes, int n_in,
                              void* d_out, int out_size, void* d_ws, size_t ws_size,
                              hipStream_t stream)
{
    (void)in_sizes; (void)n_in; (void)out_size; (void)ws_size;
    const float* x       = (const float*)d_in[0];   // (B,L,D)
    const float* alpha_p = (const float*)d_in[1];   // scalar
    const float* pos_emb = (const float*)d_in[2];   // (K,D)
    const float* w1      = (const float*)d_in[3];   // (2D,H)
    const float* b1      = (const float*)d_in[4];   // (H)
    const float* w2      = (const float*)d_in[5];   // (H,D)
    const float* b2      = (const float*)d_in[6];   // (D)

    float* x_out    = (float*)d_out;                           // (B,L,D)
    float* out_tail = x_out + (size_t)B_DIM * L_DIM * D_DIM;   // 4 scalars

    // Workspace layout (floats)
    float* ws   = (float*)d_ws;
    float* Y    = ws;                                   // (B*L, H)     262144
    float* CC   = Y    + (size_t)B_DIM * L_DIM * H_DIM; // (64, H)        8192
    float* Hbar = CC   + (size_t)64 * H_DIM;            // (M2PAD, H)   249856
    float* MD0  = Hbar + (size_t)M2PAD * H_DIM;         // (M2PAD, D)  1499136
    float* RS   = MD0  + (size_t)M2PAD * D_DIM;         // (M2PAD, 4)     7808

    // 1) Y = X * W1[:D,:]           (2048 x 768 x 128, fp32 WMMA, 32x32/wave)
    wmma_gemm_f32_kernel<<<dim3((B_DIM * L_DIM) / 32, H_DIM / 32), 32, 0, stream>>>(
        x, w1, Y, H_DIM, D_DIM, B_DIM * L_DIM);

    // 2) CC = pos_emb[:63] * W1[D:,:]   (64-padded x 768 x 128)
    wmma_gemm_f32_kernel<<<dim3(64 / 32, H_DIM / 32), 32, 0, stream>>>(
        pos_emb, w1 + (size_t)D_DIM * H_DIM, CC, H_DIM, D_DIM, K_WIN - 1);

    // 3) Hbar = windowed relu-mean over Y + CC + b1
    window_relu_mean_kernel<<<M2PAD, H_DIM, 0, stream>>>(Y, CC, b1, Hbar);

    // 4) MD0 = Hbar * W2            (1952 x 128 x 768, fp32 WMMA, 32x32/wave)
    wmma_gemm_f32_kernel<<<dim3(M2PAD / 32, D_DIM / 32), 32, 0, stream>>>(
        Hbar, w2, MD0, D_DIM, H_DIM, NROWS);

    // 5) Epilogue: updated rows of x_out + per-row stats
    epilogue_kernel<<<NROWS, 256, 0, stream>>>(x, MD0, b2, alpha_p, x_out, RS);

    // 6) Copy untouched prefix rows
    copy_head_kernel<<<(B_DIM * (K_WIN - 1) * D_DIM + 255) / 256, 256, 0, stream>>>(x, x_out);

    // 7) Deterministic scalar reduction
    finalize_kernel<<<1, 256, 0, stream>>>(RS, alpha_p, out_tail);
}